// TTM_76622216561049
// MI455X (gfx1250) — compile-verified
//
#include <hip/hip_runtime.h>
#include <math.h>

// Problem constants (match reference)
#define AVGF 128
#define SEG  64
#define MDIM 1024
#define HA   8
#define DH   128
#define TKB  4
#define LNEPS 1e-5f

#define KC  32   // K-chunk staged per double-buffer step
#define KCP 36   // padded LDS row stride (floats) to avoid bank conflicts

typedef float v2f __attribute__((ext_vector_type(2)));
typedef float v8f __attribute__((ext_vector_type(8)));

// ---------------------------------------------------------------------------
// Stage 0: segment average.  altx[f, d1*32+d2] = mean_s x[d2, d1, f*64+s]
// ---------------------------------------------------------------------------
__global__ __launch_bounds__(256) void avg_kernel(const float* __restrict__ x,
                                                  float* __restrict__ altx) {
  int idx = blockIdx.x * 256 + threadIdx.x;      // 0 .. 128*1024-1
  int f  = idx >> 10;
  int m  = idx & 1023;
  int d1 = m >> 5, d2 = m & 31;
  const float4* p = (const float4*)(x + (size_t)d2 * 32 * 8192 +
                                        (size_t)d1 * 8192 + (size_t)f * 64);
  float s = 0.f;
#pragma unroll
  for (int i = 0; i < 16; ++i) { float4 v = p[i]; s += v.x + v.y + v.z + v.w; }
  altx[idx] = s * (1.f / 64.f);
}

// ---------------------------------------------------------------------------
// WMMA GEMM:  D[row, n] = sum_k A[row,k] * W[n,k]  (+ epilogue)
//   A: [128 x K] row-major,  W: [N x K] row-major (B = W^T)
// Workgroup = 4 waves (128 thr). Each workgroup computes all 128 rows for a
// 64-col supertile; wave w owns rows [32w, 32w+32) (two 16-row tiles) x 64
// cols (four 16-col tiles) -> 8 f32 WMMA accumulators.
// A & B chunks are staged in LDS with double-buffered async copies
// (global_load_async_to_lds_b128 / s_wait_asynccnt) so WMMAs overlap DMA.
// grid = (N/64), block = 128.
// EPI: 0 none | 1 +C residual | 2 +bias | 3 +bias then exact GELU | 4 +pos_bias
// ---------------------------------------------------------------------------
template <int EPI>
__global__ __launch_bounds__(128) void wmma_gemm(const float* __restrict__ A,
                                                 const float* __restrict__ W,
                                                 const float* __restrict__ C,
                                                 const float* __restrict__ bias,
                                                 float* __restrict__ D,
                                                 int K, int N) {
  __shared__ float lA[2][128 * KCP];   // 2 x 18 KB
  __shared__ float lB[2][64 * KCP];    // 2 x  9 KB

  const int tid  = threadIdx.x;
  const int lane = tid & 31;
  const int wave = tid >> 5;
  const int half = lane >> 4;          // 0: lanes 0-15, 1: lanes 16-31
  const int r    = lane & 15;
  const int nBase = blockIdx.x << 6;
  const int rowW  = wave << 5;         // this wave's 32-row slice

  const float* gW = W + (size_t)nBase * K;     // 64 x K tile of W

  const unsigned ldsA[2] = {(unsigned)(uintptr_t)&lA[0][0],
                            (unsigned)(uintptr_t)&lA[1][0]};
  const unsigned ldsB[2] = {(unsigned)(uintptr_t)&lB[0][0],
                            (unsigned)(uintptr_t)&lB[1][0]};

  // Issue one K-chunk of async global->LDS copies (12 b128 ops per thread).
  auto issue = [&](int k0, int buf) {
    // A: 128 rows x 32 floats = 1024 x 16B -> 8 per thread, coalesced.
#pragma unroll
    for (int i = 0; i < 8; ++i) {
      int lin = i * 128 + tid;
      int row = lin >> 3;
      int q   = lin & 7;
      const float* src = A + (size_t)row * K + k0 + q * 4;
      unsigned dst = ldsA[buf] + (unsigned)(row * KCP + q * 4) * 4u;
      asm volatile("global_load_async_to_lds_b128 %0, %1, off"
                   :: "v"(dst), "v"(src) : "memory");
    }
    // B: 64 rows x 32 floats = 512 x 16B -> 4 per thread, coalesced.
#pragma unroll
    for (int i = 0; i < 4; ++i) {
      int lin = i * 128 + tid;
      int row = lin >> 3;
      int q   = lin & 7;
      const float* src = gW + (size_t)row * K + k0 + q * 4;
      unsigned dst = ldsB[buf] + (unsigned)(row * KCP + q * 4) * 4u;
      asm volatile("global_load_async_to_lds_b128 %0, %1, off"
                   :: "v"(dst), "v"(src) : "memory");
    }
  };

  v8f acc[2][4];
#pragma unroll
  for (int rt = 0; rt < 2; ++rt)
#pragma unroll
    for (int t = 0; t < 4; ++t) acc[rt][t] = (v8f){};

  const int nch = K / KC;
  issue(0, 0);
  for (int c = 0; c < nch; ++c) {
    const int buf = c & 1;
    if (c + 1 < nch) {
      issue((c + 1) * KC, buf ^ 1);
      // wait for the OLDER batch only (12 ops of the newer batch outstanding)
      asm volatile("s_wait_asynccnt 12" ::: "memory");
    } else {
      asm volatile("s_wait_asynccnt 0" ::: "memory");
    }
    __syncthreads();

#pragma unroll
    for (int ks = 0; ks < KC / 4; ++ks) {
      const int kk = ks * 4 + 2 * half;
      // fp32 16x4 A fragments: lane holds A[r, kk..kk+1]
      v2f a0 = *(const v2f*)&lA[buf][(rowW + r) * KCP + kk];
      v2f a1 = *(const v2f*)&lA[buf][(rowW + 16 + r) * KCP + kk];
      // fp32 4x16 B fragments: lane holds W[nTile + r, kk..kk+1]
      v2f b0 = *(const v2f*)&lB[buf][(r) * KCP + kk];
      v2f b1 = *(const v2f*)&lB[buf][(16 + r) * KCP + kk];
      v2f b2 = *(const v2f*)&lB[buf][(32 + r) * KCP + kk];
      v2f b3 = *(const v2f*)&lB[buf][(48 + r) * KCP + kk];
      acc[0][0] = __builtin_amdgcn_wmma_f32_16x16x4_f32(false, a0, false, b0, (short)0, acc[0][0], false, false);
      acc[0][1] = __builtin_amdgcn_wmma_f32_16x16x4_f32(false, a0, false, b1, (short)0, acc[0][1], false, false);
      acc[0][2] = __builtin_amdgcn_wmma_f32_16x16x4_f32(false, a0, false, b2, (short)0, acc[0][2], false, false);
      acc[0][3] = __builtin_amdgcn_wmma_f32_16x16x4_f32(false, a0, false, b3, (short)0, acc[0][3], false, false);
      acc[1][0] = __builtin_amdgcn_wmma_f32_16x16x4_f32(false, a1, false, b0, (short)0, acc[1][0], false, false);
      acc[1][1] = __builtin_amdgcn_wmma_f32_16x16x4_f32(false, a1, false, b1, (short)0, acc[1][1], false, false);
      acc[1][2] = __builtin_amdgcn_wmma_f32_16x16x4_f32(false, a1, false, b2, (short)0, acc[1][2], false, false);
      acc[1][3] = __builtin_amdgcn_wmma_f32_16x16x4_f32(false, a1, false, b3, (short)0, acc[1][3], false, false);
    }
    __syncthreads();   // protect LDS buffer before it is refilled
  }

  // C/D layout: VGPR j holds row (tileRow + j + 8*half), col (nTile + r)
  auto store_tile = [&](const v8f& a, int rt, int t) {
    const int col = nBase + t * 16 + r;
#pragma unroll
    for (int j = 0; j < 8; ++j) {
      const int row = rowW + rt * 16 + j + 8 * half;
      float v = a[j];
      if (EPI == 1) v += C[(size_t)row * N + col];
      if (EPI == 2 || EPI == 3) v += bias[col];
      if (EPI == 3) v = 0.5f * v * (1.f + erff(v * 0.70710678118654752f));
      if (EPI == 4) {
        int jj = col >> 1;   // pair index; even col -> sin, odd col -> cos
        float ang = (float)row * powf(10000.f, -(float)jj * (1.f / 256.f));
        v += (col & 1) ? cosf(ang) : sinf(ang);
      }
      D[(size_t)row * N + col] = v;
    }
  };
#pragma unroll
  for (int rt = 0; rt < 2; ++rt)
#pragma unroll
    for (int t = 0; t < 4; ++t) store_tile(acc[rt][t], rt, t);
}

// ---------------------------------------------------------------------------
// Row LayerNorm over 1024 cols. One block of 256 threads per row.
// ADD=true emits  Y = LN(X) + X  (in-place safe).
// ---------------------------------------------------------------------------
template <bool ADD>
__global__ __launch_bounds__(256) void ln_kernel(const float* __restrict__ X,
                                                 const float* __restrict__ g,
                                                 const float* __restrict__ b,
                                                 float* __restrict__ Y) {
  __shared__ float red[256];
  const int row = blockIdx.x;
  const float* xr = X + (size_t)row * MDIM;
  float vals[4];
  float s = 0.f;
#pragma unroll
  for (int i = 0; i < 4; ++i) { vals[i] = xr[threadIdx.x + 256 * i]; s += vals[i]; }
  red[threadIdx.x] = s;
  __syncthreads();
  for (int off = 128; off > 0; off >>= 1) {
    if (threadIdx.x < off) red[threadIdx.x] += red[threadIdx.x + off];
    __syncthreads();
  }
  const float mu = red[0] * (1.f / 1024.f);
  __syncthreads();
  float s2 = 0.f;
#pragma unroll
  for (int i = 0; i < 4; ++i) { float d = vals[i] - mu; s2 += d * d; }
  red[threadIdx.x] = s2;
  __syncthreads();
  for (int off = 128; off > 0; off >>= 1) {
    if (threadIdx.x < off) red[threadIdx.x] += red[threadIdx.x + off];
    __syncthreads();
  }
  const float inv = rsqrtf(red[0] * (1.f / 1024.f) + LNEPS);
#pragma unroll
  for (int i = 0; i < 4; ++i) {
    int c = threadIdx.x + 256 * i;
    float ln = (vals[i] - mu) * inv * g[c] + b[c];
    Y[(size_t)row * MDIM + c] = ADD ? (ln + vals[i]) : ln;
  }
}

// ---------------------------------------------------------------------------
// No-softmax attention: rsa[i,h] = (q.k)/sqrt(DH); cum = cumsum_i(rsa*v).
// qkv: [128 x 3072] (q | k | v per row, col = h*128+d). One 1024-thread block.
// ---------------------------------------------------------------------------
__global__ __launch_bounds__(1024) void attn_kernel(const float* __restrict__ qkv,
                                                    float* __restrict__ cum) {
  __shared__ float rsa[AVGF][HA];
  const int tid = threadIdx.x;
  {
    const int i = tid >> 3;
    const int h = tid & 7;
    const float4* q4 = (const float4*)(qkv + (size_t)i * 3072 + h * DH);
    const float4* k4 = (const float4*)(qkv + (size_t)i * 3072 + MDIM + h * DH);
    float s = 0.f;
#pragma unroll 4
    for (int d = 0; d < DH / 4; ++d) {
      float4 qa = q4[d], ka = k4[d];
      s += qa.x * ka.x + qa.y * ka.y + qa.z * ka.z + qa.w * ka.w;
    }
    rsa[i][h] = s * 0.08838834764831845f;  // 1/sqrt(128)
  }
  __syncthreads();
  const int m = tid;          // column 0..1023
  const int h = m >> 7;
  float acc = 0.f;
  for (int i = 0; i < AVGF; ++i) {
    acc += rsa[i][h] * qkv[(size_t)i * 3072 + 2048 + m];
    cum[(size_t)i * MDIM + m] = acc;
  }
}

// ---------------------------------------------------------------------------
extern "C" void kernel_launch(void* const* d_in, const int* in_sizes, int n_in,
                              void* d_out, int out_size, void* d_ws, size_t ws_size,
                              hipStream_t stream) {
  const float* x      = (const float*)d_in[0];
  const float* weight = (const float*)d_in[1];
  const float* Wqkv   = (const float*)d_in[2];
  const float* Wo     = (const float*)d_in[3];
  const float* ln1_g  = (const float*)d_in[4];
  const float* ln1_b  = (const float*)d_in[5];
  const float* ln2_g  = (const float*)d_in[6];
  const float* ln2_b  = (const float*)d_in[7];
  const float* fc1_w  = (const float*)d_in[8];
  const float* fc1_b  = (const float*)d_in[9];
  const float* fc2_w  = (const float*)d_in[10];
  const float* fc2_b  = (const float*)d_in[11];
  float* out = (float*)d_out;

  float* ws   = (float*)d_ws;
  float* altx = ws;                       // 128*1024
  float* save = altx + 128 * 1024;        // 128*1024
  float* z    = save + 128 * 1024;        // 128*1024
  float* qkv  = z    + 128 * 1024;        // 128*3072
  float* cum  = qkv  + 128 * 3072;        // 128*1024
  float* hbuf = cum  + 128 * 1024;        // 128*4096

  avg_kernel<<<512, 256, 0, stream>>>(x, altx);

  // save = altx @ weight^T + pos_bias
  wmma_gemm<4><<<16, 128, 0, stream>>>(altx, weight, nullptr, nullptr,
                                       save, 1024, 1024);

  for (int a = 0; a < TKB; ++a) {
    const float* Wqkv_a = Wqkv + (size_t)a * 3 * 1024 * 1024;
    const float* Wo_a   = Wo   + (size_t)a * 1024 * 1024;

    // z = LN1(save)
    ln_kernel<false><<<128, 256, 0, stream>>>(save, ln1_g, ln1_b, z);
    // qkv = z @ Wqkv^T  (N=3072)
    wmma_gemm<0><<<48, 128, 0, stream>>>(z, Wqkv_a, nullptr, nullptr,
                                         qkv, 1024, 3072);
    // rsa + scaled cumsum
    attn_kernel<<<1, 1024, 0, stream>>>(qkv, cum);
    // save = cum @ Wo^T + save
    wmma_gemm<1><<<16, 128, 0, stream>>>(cum, Wo_a, save, nullptr,
                                         save, 1024, 1024);
    // save = LN2(save) + save
    ln_kernel<true><<<128, 256, 0, stream>>>(save, ln2_g, ln2_b, save);
    // hbuf = gelu(save @ fc1^T + b1)   (N=4096)
    wmma_gemm<3><<<64, 128, 0, stream>>>(save, fc1_w, nullptr, fc1_b,
                                         hbuf, 1024, 4096);
    // save/out = hbuf @ fc2^T + b2   (K=4096)
    float* dst = (a == TKB - 1) ? out : save;
    wmma_gemm<2><<<16, 128, 0, stream>>>(hbuf, fc2_w, nullptr, fc2_b,
                                         dst, 4096, 1024);
  }
}